// gconvnet_regression_confounded_75505525063924
// MI455X (gfx1250) — compile-verified
//
#include <hip/hip_runtime.h>
#include <hip/hip_bf16.h>

typedef __attribute__((ext_vector_type(2))) float v2f;
typedef __attribute__((ext_vector_type(8))) float v8f;

#define NGRAPH 4

static __device__ __forceinline__ unsigned keyf(float s) {
    unsigned u = __float_as_uint(s);
    return (u & 0x80000000u) ? ~u : (u | 0x80000000u);
}

// ---------------- GEMM via V_WMMA_F32_16X16X4_F32 ----------------
// One wave (32 lanes) computes one 16x16 tile of H = X[M,K] @ W[K,N].
__global__ void gemm_wmma(const float* __restrict__ X, const float* __restrict__ W,
                          float* __restrict__ H, int M, int Kd, int Nf) {
    const int tn = blockIdx.x;            // tile col
    const int tm = blockIdx.y;            // tile row
    const int lane = threadIdx.x;         // 0..31
    const int half = lane >> 4;           // 0/1
    const int l16  = lane & 15;
    int rowA = tm * 16 + l16;
    int rowAc = rowA < M ? rowA : (M - 1);      // clamp loads; stores predicated
    const int colB = tn * 16 + l16;
    v8f acc = {0.f, 0.f, 0.f, 0.f, 0.f, 0.f, 0.f, 0.f};
    for (int kk = 0; kk < Kd; kk += 4) {
        int k0 = kk + half * 2;
        v2f a, b;
        a.x = X[(size_t)rowAc * Kd + k0];
        a.y = X[(size_t)rowAc * Kd + k0 + 1];
        b.x = W[(size_t)k0 * Nf + colB];
        b.y = W[(size_t)(k0 + 1) * Nf + colB];
        acc = __builtin_amdgcn_wmma_f32_16x16x4_f32(false, a, false, b,
                                                    (short)0, acc, false, false);
    }
    for (int r = 0; r < 8; ++r) {
        int m = tm * 16 + r + 8 * half;
        if (m < M) H[(size_t)m * Nf + colB] = acc[r];
    }
}

// ---------------- small utility kernels ----------------
__global__ void fill_f32(float* p, float v, int n) {
    int i = blockIdx.x * blockDim.x + threadIdx.x;
    if (i < n) p[i] = v;
}
__global__ void fill_i32(int* p, int v, int n) {
    int i = blockIdx.x * blockDim.x + threadIdx.x;
    if (i < n) p[i] = v;
}
__global__ void init_edges(const int* __restrict__ ei, int* src, int* dst, float* w, int E) {
    int e = blockIdx.x * blockDim.x + threadIdx.x;
    if (e >= E) return;
    src[e] = ei[e];
    dst[e] = ei[E + e];
    w[e] = 1.0f;
}

// ---------------- GCN pieces ----------------
__global__ void deg_acc(const int* __restrict__ dst, const float* __restrict__ w,
                        float* deg, int E) {
    int e = blockIdx.x * blockDim.x + threadIdx.x;
    if (e >= E) return;
    float we = w[e];
    if (we != 0.f) atomicAdd(&deg[dst[e]], we);
}
__global__ void rsqrt_k(const float* deg, float* dinv, int M) {
    int i = blockIdx.x * blockDim.x + threadIdx.x;
    if (i < M) dinv[i] = rsqrtf(deg[i]);
}
__global__ void agg_edges(const float* __restrict__ h, const int* __restrict__ src,
                          const int* __restrict__ dst, const float* __restrict__ w,
                          const float* __restrict__ dinv, float* agg,
                          int E, int F, int chunks) {
    int idx = blockIdx.x * blockDim.x + threadIdx.x;
    if (idx >= E * chunks) return;
    int e = idx / chunks;
    float we = w[e];
    if (we == 0.f) return;
    int c = (idx - e * chunks) * 4;
    int s = src[e], d = dst[e];
    float nrm = dinv[s] * dinv[d] * we;
    const float* hs = h + (size_t)s * F + c;
    float* ad = agg + (size_t)d * F + c;
    atomicAdd(ad + 0, hs[0] * nrm);
    atomicAdd(ad + 1, hs[1] * nrm);
    atomicAdd(ad + 2, hs[2] * nrm);
    atomicAdd(ad + 3, hs[3] * nrm);
}
__global__ void combine_relu(float* agg, const float* __restrict__ h,
                             const float* __restrict__ dinv, const float* __restrict__ bias,
                             int M, int F) {
    int idx = blockIdx.x * blockDim.x + threadIdx.x;
    if (idx >= M * F) return;
    int i = idx / F, f = idx - i * F;
    float di = dinv[i];
    float v = agg[idx] + h[idx] * di * di + bias[f];
    agg[idx] = fmaxf(v, 0.f);
}

// ---------------- TopK pooling ----------------
__global__ void pnorm_k(const float* __restrict__ p, int F, float* invpn) {
    __shared__ float sh[128];
    int t = threadIdx.x;
    float v = (t < F) ? p[t] * p[t] : 0.f;
    sh[t] = v;
    __syncthreads();
    for (int st = 64; st > 0; st >>= 1) {
        if (t < st) sh[t] += sh[t + st];
        __syncthreads();
    }
    if (t == 0) *invpn = rsqrtf(sh[0]);
}
__global__ void score_k(const float* __restrict__ x, const float* __restrict__ p,
                        const float* __restrict__ invpn, float* score, int M, int F) {
    int i = blockIdx.x * blockDim.x + threadIdx.x;
    if (i >= M) return;
    const float* xi = x + (size_t)i * F;
    float s = 0.f;
    for (int f = 0; f < F; ++f) s += xi[f] * p[f];
    score[i] = s * (*invpn);
}
__global__ void sel_init(unsigned* prefix, int* remain, int* slotc, int* tiec, int k) {
    int b = threadIdx.x;
    if (b >= NGRAPH) return;
    prefix[b] = 0u;
    remain[b] = k;
    slotc[b] = 0;
    tiec[b] = 0;
}
__global__ void hist_k(const float* __restrict__ score, const unsigned* __restrict__ prefix,
                       int* hist, int M, int n, int pass) {
    int i = blockIdx.x * blockDim.x + threadIdx.x;
    if (i >= M) return;
    int b = i / n;
    unsigned u = keyf(score[i]);
    bool ok;
    if (pass == 3) ok = true;
    else {
        int sh = 8 * pass + 8;
        ok = ((u >> sh) == (prefix[b] >> sh));
    }
    if (ok) atomicAdd(&hist[(b << 8) + ((u >> (8 * pass)) & 255u)], 1);
}
__global__ void scan_k(unsigned* prefix, int* remain, const int* __restrict__ hist, int pass) {
    int b = threadIdx.x;
    if (b >= NGRAPH) return;
    int r = remain[b];
    unsigned pref = prefix[b];
    int acc = 0;
    for (int bin = 255; bin >= 0; --bin) {
        int c = hist[(b << 8) + bin];
        if (acc + c >= r) {
            prefix[b] = pref | ((unsigned)bin << (8 * pass));
            remain[b] = r - acc;
            return;
        }
        acc += c;
    }
}
__global__ void select_k(const float* __restrict__ x, const float* __restrict__ score,
                         const unsigned* __restrict__ thresh, const int* __restrict__ need_eq,
                         int* slotc, int* tiec, int* remap, float* xn,
                         int M, int n, int k, int F) {
    int i = blockIdx.x * blockDim.x + threadIdx.x;
    if (i >= M) return;
    int b = i / n;
    float s = score[i];
    unsigned u = keyf(s);
    unsigned T = thresh[b];
    int nid = -1;
    if (u > T) {
        nid = atomicAdd(&slotc[b], 1);
    } else if (u == T) {
        int t = atomicAdd(&tiec[b], 1);
        if (t < need_eq[b]) nid = atomicAdd(&slotc[b], 1);
    }
    if (nid >= 0) {
        int g = b * k + nid;
        remap[i] = g;
        float sc = tanhf(s);
        const float* xi = x + (size_t)i * F;
        float* xo = xn + (size_t)g * F;
        for (int f = 0; f < F; ++f) xo[f] = xi[f] * sc;
    } else {
        remap[i] = -1;
    }
}
__global__ void remap_edges(int* src, int* dst, float* w, const int* __restrict__ remap, int E) {
    int e = blockIdx.x * blockDim.x + threadIdx.x;
    if (e >= E) return;
    int ns = remap[src[e]];
    int nd = remap[dst[e]];
    bool valid = (ns >= 0) && (nd >= 0);
    src[e] = valid ? ns : 0;
    dst[e] = valid ? nd : 0;
    w[e] = valid ? w[e] : 0.f;
}

// ---------------- readout ----------------
__global__ void readout(const float* __restrict__ x, const float* __restrict__ metadata,
                        const float* __restrict__ convm_w, const float* __restrict__ convm_b,
                        const float* __restrict__ fc_w, const float* __restrict__ fc_b,
                        const float* __restrict__ fc2_w, const float* __restrict__ fc2_b,
                        float* out, int k, int F) {
    __shared__ float xc[260];
    __shared__ float hh[128];
    int b = blockIdx.x;
    int f = threadIdx.x;  // 0..127
    {
        float mx = -3.402823e38f, sm = 0.f;
        const float* xb = x + ((size_t)b * k) * F + f;
        for (int j = 0; j < k; ++j) {
            float v = xb[(size_t)j * F];
            mx = fmaxf(mx, v);
            sm += v;
        }
        xc[f] = mx;
        xc[F + f] = sm / (float)k;
    }
    if (f < 4) {
        float mv = metadata[b] * convm_w[f] + convm_b[f];
        xc[2 * F + f] = fmaxf(mv, 0.f);
    }
    __syncthreads();
    {
        float a = fc_b[f];
        for (int i = 0; i < 2 * F + 4; ++i) a += xc[i] * fc_w[i * F + f];
        hh[f] = fmaxf(a, 0.f) * fc2_w[f];
    }
    __syncthreads();
    for (int st = 64; st > 0; st >>= 1) {
        if (f < st) hh[f] += hh[f + st];
        __syncthreads();
    }
    if (f == 0) out[b] = hh[0] + fc2_b[0];
}

static inline int nblk(int n) { return (n + 255) / 256; }

extern "C" void kernel_launch(void* const* d_in, const int* in_sizes, int n_in,
                              void* d_out, int out_size, void* d_ws, size_t ws_size,
                              hipStream_t stream) {
    const float* x0       = (const float*)d_in[0];
    const int*   ei       = (const int*)d_in[1];
    const float* metadata = (const float*)d_in[2];
    const float* convm_w  = (const float*)d_in[15];
    const float* convm_b  = (const float*)d_in[16];
    const float* fc_w     = (const float*)d_in[17];
    const float* fc_b     = (const float*)d_in[18];
    const float* fc2_w    = (const float*)d_in[19];
    const float* fc2_b    = (const float*)d_in[20];

    const int E  = in_sizes[1] / 2;        // 983088
    const int M0 = in_sizes[0] / 4;        // 163848

    static const int Fin[4]  = {4, 32, 64, 128};
    static const int Fout[4] = {32, 64, 128, 128};
    static const int Kk[4]   = {20481, 10241, 5121, 2561};
    const int Ms[4] = {M0, NGRAPH * Kk[0], NGRAPH * Kk[1], NGRAPH * Kk[2]};

    // workspace carve-up (all f32/i32, 4-byte aligned)
    const size_t HMAX = 5300000;   // >= max M*Fout
    const size_t XMAX = 2700000;   // >= max pooled B*k*F
    const size_t MMAX = 164096;    // >= M0
    float* ws = (float*)d_ws;
    float*    hbuf  = ws;               ws += HMAX;
    float*    aggb  = ws;               ws += HMAX;
    float*    xb    = ws;               ws += XMAX;
    float*    deg   = ws;               ws += MMAX;
    float*    dinv  = ws;               ws += MMAX;
    float*    score = ws;               ws += MMAX;
    int*      remap = (int*)ws;         ws += MMAX;
    int*      srcb  = (int*)ws;         ws += (size_t)E;
    int*      dstb  = (int*)ws;         ws += (size_t)E;
    float*    wts   = ws;               ws += (size_t)E;
    int*      hist  = (int*)ws;         ws += NGRAPH * 256;
    unsigned* prefix= (unsigned*)ws;    ws += NGRAPH;
    int*      remain= (int*)ws;         ws += NGRAPH;
    int*      slotc = (int*)ws;         ws += NGRAPH;
    int*      tiec  = (int*)ws;         ws += NGRAPH;
    float*    invpn = ws;               ws += 1;

    init_edges<<<nblk(E), 256, 0, stream>>>(ei, srcb, dstb, wts, E);

    const float* xin = x0;
    for (int i = 0; i < 4; ++i) {
        const int M = Ms[i], n = M / NGRAPH, k = Kk[i];
        const int Fi = Fin[i], Fo = Fout[i];
        const float* W    = (const float*)d_in[3 + 3 * i];
        const float* bias = (const float*)d_in[4 + 3 * i];
        const float* p    = (const float*)d_in[5 + 3 * i];

        // h = x @ W   (WMMA f32)
        dim3 gg(Fo / 16, (M + 15) / 16);
        gemm_wmma<<<gg, 32, 0, stream>>>(xin, W, hbuf, M, Fi, Fo);

        // degrees (self loop = 1) and rsqrt
        fill_f32<<<nblk(M), 256, 0, stream>>>(deg, 1.0f, M);
        deg_acc<<<nblk(E), 256, 0, stream>>>(dstb, wts, deg, E);
        rsqrt_k<<<nblk(M), 256, 0, stream>>>(deg, dinv, M);

        // agg = segment_sum(h[src] * norm)
        const int MF = M * Fo;
        fill_f32<<<nblk(MF), 256, 0, stream>>>(aggb, 0.f, MF);
        const int chunks = Fo / 4;
        agg_edges<<<nblk(E * chunks), 256, 0, stream>>>(hbuf, srcb, dstb, wts, dinv, aggb,
                                                        E, Fo, chunks);
        combine_relu<<<nblk(MF), 256, 0, stream>>>(aggb, hbuf, dinv, bias, M, Fo);

        // TopK pooling (ratio 0.5 -> k per graph)
        pnorm_k<<<1, 128, 0, stream>>>(p, Fo, invpn);
        score_k<<<nblk(M), 256, 0, stream>>>(aggb, p, invpn, score, M, Fo);
        sel_init<<<1, 32, 0, stream>>>(prefix, remain, slotc, tiec, k);
        for (int pass = 3; pass >= 0; --pass) {
            fill_i32<<<nblk(NGRAPH * 256), 256, 0, stream>>>(hist, 0, NGRAPH * 256);
            hist_k<<<nblk(M), 256, 0, stream>>>(score, prefix, hist, M, n, pass);
            scan_k<<<1, 32, 0, stream>>>(prefix, remain, hist, pass);
        }
        select_k<<<nblk(M), 256, 0, stream>>>(aggb, score, prefix, remain, slotc, tiec,
                                              remap, xb, M, n, k, Fo);
        remap_edges<<<nblk(E), 256, 0, stream>>>(srcb, dstb, wts, remap, E);
        xin = xb;
    }

    readout<<<NGRAPH, 128, 0, stream>>>(xb, metadata, convm_w, convm_b,
                                        fc_w, fc_b, fc2_w, fc2_b,
                                        (float*)d_out, Kk[3], 128);
}